// Self_L1_53927609369071
// MI455X (gfx1250) — compile-verified
//
#include <hip/hip_runtime.h>
#include <hip/hip_bf16.h>

typedef __attribute__((ext_vector_type(2))) float v2f;
typedef __attribute__((ext_vector_type(8))) float v8f;

#define NB    8192
#define NC    128
#define NEGL  9.210340371976184f   /* -log(1e-4) */
#define CMIN  1e-4f

__global__ void kl_zero_acc(double* acc) {
    if (threadIdx.x == 0) acc[0] = 0.0;
}

// One wave (32 lanes) per row; each lane owns 4 consecutive columns.
__global__ void kl_softmax_rows(const float* __restrict__ s,
                                const int*   __restrict__ t,
                                const float* __restrict__ cw,
                                float* __restrict__ P,
                                float* __restrict__ LP,
                                float* __restrict__ R,
                                float* __restrict__ WT,
                                int*   __restrict__ TT) {
    const int wave = threadIdx.x >> 5;
    const int lane = threadIdx.x & 31;
    const int row  = blockIdx.x * 8 + wave;

    const float4 v = reinterpret_cast<const float4*>(s + (size_t)row * NC)[lane];

    float m = fmaxf(fmaxf(v.x, v.y), fmaxf(v.z, v.w));
    #pragma unroll
    for (int off = 16; off > 0; off >>= 1)
        m = fmaxf(m, __shfl_xor(m, off, 32));

    float4 e;
    e.x = expf(v.x - m); e.y = expf(v.y - m);
    e.z = expf(v.z - m); e.w = expf(v.w - m);
    float sum = e.x + e.y + e.z + e.w;
    #pragma unroll
    for (int off = 16; off > 0; off >>= 1)
        sum += __shfl_xor(sum, off, 32);

    const float lz = logf(sum);
    float4 lp, p;
    lp.x = v.x - m - lz; lp.y = v.y - m - lz;
    lp.z = v.z - m - lz; lp.w = v.w - m - lz;
    p.x = expf(lp.x); p.y = expf(lp.y); p.z = expf(lp.z); p.w = expf(lp.w);

    float r = p.x * lp.x + p.y * lp.y + p.z * lp.z + p.w * lp.w;
    #pragma unroll
    for (int off = 16; off > 0; off >>= 1)
        r += __shfl_xor(r, off, 32);

    reinterpret_cast<float4*>(P  + (size_t)row * NC)[lane] = p;
    reinterpret_cast<float4*>(LP + (size_t)row * NC)[lane] = lp;
    if (lane == 0) {
        R[row] = r;
        const int tr = t[row];
        TT[row] = tr;
        WT[row] = cw[tr];
    }
}

// 8 waves/block arranged 2(i) x 4(j); each wave owns a 64x64 output tile as
// 4x4 V_WMMA_F32_16X16X4_F32 accumulators (128 VGPRs), 8 loads : 16 wmma per
// k-step. Tile displacements live in the 24-bit instruction offset (u*8192 B).
// __launch_bounds__(256,1) lifts the default 128-VGPR occupancy cap so the
// accumulators stay resident (no scratch spills).
__global__ void __launch_bounds__(256, 1)
kl_pairwise(const float* __restrict__ P,
            const float* __restrict__ LP,
            const float* __restrict__ R,
            const float* __restrict__ WT,
            const int*   __restrict__ TT,
            double* __restrict__ acc) {
    const int wave = threadIdx.x >> 5;
    const int lane = threadIdx.x & 31;
    const int wi = wave & 1;          // 2 waves along i
    const int wj = wave >> 1;         // 4 waves along j
    const int i0 = blockIdx.y * 128 + wi * 64;
    const int j0 = blockIdx.x * 256 + wj * 64;

    const int kh  = lane >> 4;        // K-half select (ISA 16x16x4 f32 layout)
    const int idx = lane & 15;        // M (for A) / N (for B)

    const float* Abase = P  + (size_t)(i0 + idx) * NC + kh * 2;
    const float* Bbase = LP + (size_t)(j0 + idx) * NC + kh * 2;

    v8f c[4][4];
    #pragma unroll
    for (int ii = 0; ii < 4; ++ii)
        #pragma unroll
        for (int jj = 0; jj < 4; ++jj)
            c[ii][jj] = (v8f){};

    #pragma unroll 2
    for (int k = 0; k < NC; k += 4) {
        v2f a[4], b[4];
        #pragma unroll
        for (int u = 0; u < 4; ++u) {
            a[u] = *(const v2f*)(Abase + k + u * (16 * NC));  // offset: u*8192 B
            b[u] = *(const v2f*)(Bbase + k + u * (16 * NC));
        }
        #pragma unroll
        for (int ii = 0; ii < 4; ++ii)
            #pragma unroll
            for (int jj = 0; jj < 4; ++jj)
                c[ii][jj] = __builtin_amdgcn_wmma_f32_16x16x4_f32(
                    false, a[ii], false, b[jj], (short)0, c[ii][jj], false, false);
    }

    // C/D layout: lane holds column n = (lane&15), VGPR g holds row m = (lane>>4)*8 + g.
    const int mB = kh * 8;
    int   tj[4];
    float wjv[4];
    #pragma unroll
    for (int u = 0; u < 4; ++u) {
        const int j = j0 + u * 16 + idx;
        tj[u]  = TT[j];
        wjv[u] = WT[j] * CMIN;
    }

    float part = 0.0f;
    #pragma unroll
    for (int ii = 0; ii < 4; ++ii) {
        #pragma unroll
        for (int g = 0; g < 8; ++g) {
            const int   i   = i0 + ii * 16 + mB + g;
            const float ri  = R[i];
            const int   ti  = TT[i];
            const float wiv = WT[i];
            #pragma unroll
            for (int jj = 0; jj < 4; ++jj) {
                const float a  = ri - c[ii][jj][g];
                const float bq = (ti == tj[jj]) ? 0.0f : NEGL * (wiv - wjv[jj]);
                part += fabsf(a - bq);
            }
        }
    }

    __shared__ float red[256];
    red[threadIdx.x] = part;
    __syncthreads();
    #pragma unroll
    for (int st = 128; st > 0; st >>= 1) {
        if (threadIdx.x < st) red[threadIdx.x] += red[threadIdx.x + st];
        __syncthreads();
    }
    if (threadIdx.x == 0) atomicAdd(acc, (double)red[0]);
}

__global__ void kl_finalize(const double* __restrict__ acc, float* __restrict__ out) {
    if (threadIdx.x == 0)
        out[0] = (float)((1.0e-4 + acc[0]) / ((double)NB * (double)NB));
}

extern "C" void kernel_launch(void* const* d_in, const int* in_sizes, int n_in,
                              void* d_out, int out_size, void* d_ws, size_t ws_size,
                              hipStream_t stream) {
    const float* s  = (const float*)d_in[0];
    const int*   t  = (const int*)  d_in[1];
    const float* cw = (const float*)d_in[2];

    float* P  = (float*)d_ws;                        // [NB*NC]
    float* LP = P  + (size_t)NB * NC;                // [NB*NC]
    float* R  = LP + (size_t)NB * NC;                // [NB]
    float* WT = R  + NB;                             // [NB]
    int*   TT = (int*)(WT + NB);                     // [NB]
    double* acc = (double*)(TT + NB);                // 8-byte aligned (offset 8486912)

    kl_zero_acc<<<1, 32, 0, stream>>>(acc);
    kl_softmax_rows<<<NB / 8, 256, 0, stream>>>(s, t, cw, P, LP, R, WT, TT);
    dim3 grid(NB / 256, NB / 128);
    kl_pairwise<<<grid, 256, 0, stream>>>(P, LP, R, WT, TT, acc);
    kl_finalize<<<1, 32, 0, stream>>>(acc, (float*)d_out);
}